// ModelGAT1_1958505087518
// MI455X (gfx1250) — compile-verified
//
#include <hip/hip_runtime.h>

typedef float v2f __attribute__((ext_vector_type(2)));
typedef float v8f __attribute__((ext_vector_type(8)));
typedef unsigned v4u __attribute__((ext_vector_type(4)));
typedef int v4i __attribute__((ext_vector_type(4)));
typedef int v8i __attribute__((ext_vector_type(8)));

#define NU 200000
#define NI 100000
#define DIN 32
#define HH 64
#define EE 2000000
#define LL 500000
#define NEG_SLOPE 0.2f

#if defined(__HIP_DEVICE_COMPILE__) && __has_builtin(__builtin_amdgcn_tensor_load_to_lds)
#define HAVE_TDM 1
#else
#define HAVE_TDM 0
#endif

// ---- order-preserving float<->uint encoding for atomic max on signed floats ----
__device__ __forceinline__ unsigned fenc(float f) {
  unsigned u = __float_as_uint(f);
  return (u & 0x80000000u) ? ~u : (u | 0x80000000u);
}
__device__ __forceinline__ float fdec(unsigned u) {
  return (u & 0x80000000u) ? __uint_as_float(u & 0x7fffffffu) : __uint_as_float(~u);
}

#if HAVE_TDM
// Stage nelem f32 (contiguous) from global into LDS via the Tensor Data Mover.
// D# per CDNA5 ISA ch.8: 1-D tile (tile_dim0 = tensor_dim0 = stride0 = nelem,
// tensor_dim1 = 1, tile_dim1/2 = 0), data_size code 2 (4 bytes), type = 2.
// This toolchain's builtin is the 6-arg form: (v4u, v8i, v4i, v4i, v8i, i32 cpol).
__device__ __forceinline__ void tdm_stage_f32(const float* gsrc, void* lds_dst,
                                              unsigned nelem) {
  unsigned lds_off = (unsigned)(uintptr_t)lds_dst;  // flat LDS addr[31:0] == LDS offset
  unsigned long long ga = (unsigned long long)(uintptr_t)gsrc;
  v4u g0 = {1u,                                   // count=1, user descriptor
            lds_off,                              // lds_addr
            (unsigned)ga,                         // global_addr[31:0]
            (unsigned)((ga >> 32) & 0x01FFFFFFu) | 0x80000000u};  // ga[56:32] | type=2
  v8i g1 = {(int)(2u << 16),                      // workgroup_mask=0, data_size=4B
            (int)((nelem & 0xFFFFu) << 16),       // tensor_dim0[15:0]
            (int)((nelem >> 16) | (1u << 16)),    // tensor_dim0[31:16], tensor_dim1=1
            (int)((nelem & 0xFFFFu) << 16),       // tensor_dim1[31:16]=0, tile_dim0
            0,                                    // tile_dim1=0 (1-D), tile_dim2=0
            (int)nelem,                           // tensor_dim0_stride[31:0]
            0, 0};                                // stride hi / dim1_stride = 0
  v4i z4 = {0, 0, 0, 0};
  v8i z8 = {0, 0, 0, 0, 0, 0, 0, 0};
  __builtin_amdgcn_tensor_load_to_lds(g0, g1, z4, z4, z8, 0);
  __builtin_amdgcn_s_wait_tensorcnt(0);
}
#endif

// ================= fills =================
__global__ void k_fill_u32(unsigned* __restrict__ p, unsigned v, int n) {
  int i = blockIdx.x * blockDim.x + threadIdx.x;
  if (i < n) p[i] = v;
}
// out[node][f] = b[f]  (accumulator pre-seeded with conv bias)
__global__ void k_init_bias(float* __restrict__ out, const float* __restrict__ b, int total) {
  int i = blockIdx.x * blockDim.x + threadIdx.x;
  if (i < total) out[i] = b[i & (HH - 1)];
}

// ================= wave-per-row dot: out[r] = X[r,:K] . v (+sigmoid) =========
__global__ void k_dot_rows(const float* __restrict__ X, const float* __restrict__ v,
                           float* __restrict__ out, int n, int K,
                           int sigmoid_mode, const float* __restrict__ bias1) {
  int wid = blockIdx.x * 8 + (threadIdx.x >> 5);
  int lane = threadIdx.x & 31;
  if (wid >= n) return;
  float s = 0.f;
  for (int k = lane; k < K; k += 32) s += X[(size_t)wid * K + k] * v[k];
  for (int off = 16; off > 0; off >>= 1) s += __shfl_xor(s, off, 32);
  if (lane == 0) {
    if (sigmoid_mode) {
      float b = bias1 ? bias1[0] : 0.f;
      s = 1.f / (1.f + expf(-(s + b)));
    }
    out[wid] = s;
  }
}

// ================= WMMA GEMM: out(MxHH) = A(MxK) @ B(KxHH) [+bias +addend +relu]
// block = 256 threads = 8 waves; each wave -> one 16x16 tile; block -> 32x64.
// A f32 16x4 frag: lane m=lane&15, lane-half picks K pair, vgpr picks parity.
// B f32 4x16 frag: lane n=lane&15, same K split. C/D: m = r + 8*(lane>>4).
__global__ void __launch_bounds__(256)
k_gemm_wmma(const float* __restrict__ A, const float* __restrict__ B,
            const float* __restrict__ bias, const float* __restrict__ addend,
            float* __restrict__ out, int K, int relu) {
  __shared__ float Bs[128 * HH];
  int tid = threadIdx.x;
#if HAVE_TDM
  if (tid < 32) tdm_stage_f32(B, Bs, (unsigned)(K * HH));
#else
  for (int i = tid; i < K * HH; i += 256) Bs[i] = B[i];
#endif
  __syncthreads();
  int wave = tid >> 5, lane = tid & 31;
  int mr = lane & 15, kh = lane >> 4;
  int m_tile = blockIdx.x * 32 + (wave >> 2) * 16;
  int n_tile = (wave & 3) * 16;
  v8f c = {0.f, 0.f, 0.f, 0.f, 0.f, 0.f, 0.f, 0.f};
  const float* arow = A + (size_t)(m_tile + mr) * K;
  for (int k0 = 0; k0 < K; k0 += 4) {
    int k = k0 + (kh << 1);
    v2f a = *reinterpret_cast<const v2f*>(arow + k);  // 8B aligned: m*K, k even
    v2f b;
    b[0] = Bs[k * HH + n_tile + mr];
    b[1] = Bs[(k + 1) * HH + n_tile + mr];
    c = __builtin_amdgcn_wmma_f32_16x16x4_f32(false, a, false, b, (short)0, c, false, false);
  }
  int nn = n_tile + mr;
  int m0 = m_tile + (kh << 3);
  float bb = bias ? bias[nn] : 0.f;
#pragma unroll
  for (int r = 0; r < 8; ++r) {
    size_t o = (size_t)(m0 + r) * HH + nn;
    float v = c[r] + bb;
    if (addend) v += addend[o];
    if (relu) v = fmaxf(v, 0.f);
    out[o] = v;
  }
}

// ============ decoder GEMM with fused endpoint gather: K=128 fixed ===========
// A[m][k] = (k<64) ? z2u[rows[m]][k] : z2i[cols[m]][k-64];  out = relu(A@W + b)
__global__ void __launch_bounds__(256)
k_dec_gemm_wmma(const float* __restrict__ zu, const float* __restrict__ zi,
                const int* __restrict__ rows, const int* __restrict__ cols,
                const float* __restrict__ W, const float* __restrict__ bias,
                float* __restrict__ out) {
  __shared__ float Bs[128 * HH];
  int tid = threadIdx.x;
#if HAVE_TDM
  if (tid < 32) tdm_stage_f32(W, Bs, 128u * HH);
#else
  for (int i = tid; i < 128 * HH; i += 256) Bs[i] = W[i];
#endif
  __syncthreads();
  int wave = tid >> 5, lane = tid & 31;
  int mr = lane & 15, kh = lane >> 4;
  int m_tile = blockIdx.x * 32 + (wave >> 2) * 16;
  int n_tile = (wave & 3) * 16;
  int gm = m_tile + mr;
  const float* ru = zu + (size_t)rows[gm] * HH;
  const float* ri = zi + (size_t)cols[gm] * HH;
  v8f c = {0.f, 0.f, 0.f, 0.f, 0.f, 0.f, 0.f, 0.f};
  for (int k0 = 0; k0 < 128; k0 += 4) {
    int k = k0 + (kh << 1);
    const float* src = (k < HH) ? (ru + k) : (ri + (k - HH));
    v2f a = *reinterpret_cast<const v2f*>(src);
    v2f b;
    b[0] = Bs[k * HH + n_tile + mr];
    b[1] = Bs[(k + 1) * HH + n_tile + mr];
    c = __builtin_amdgcn_wmma_f32_16x16x4_f32(false, a, false, b, (short)0, c, false, false);
  }
  int nn = n_tile + mr;
  int m0 = m_tile + (kh << 3);
  float bb = bias[nn];
#pragma unroll
  for (int r = 0; r < 8; ++r) {
    size_t o = (size_t)(m0 + r) * HH + nn;
    out[o] = fmaxf(c[r] + bb, 0.f);
  }
}

// ================= edge kernels =================
__global__ void k_edge_e_max(const int* __restrict__ src, const int* __restrict__ dst,
                             const float* __restrict__ aS, const float* __restrict__ aD,
                             float* __restrict__ ebuf, unsigned* __restrict__ menc, int E) {
  int t = blockIdx.x * blockDim.x + threadIdx.x;
  if (t >= E) return;
  int s = src[t], d = dst[t];
  float v = aS[s] + aD[d];
  float e = v > 0.f ? v : NEG_SLOPE * v;
  ebuf[t] = e;
  atomicMax(&menc[d], fenc(e));
}

__global__ void k_edge_exp_sum(const int* __restrict__ dst, const float* __restrict__ ebuf,
                               const unsigned* __restrict__ menc, float* __restrict__ exbuf,
                               float* __restrict__ denom, int E) {
  int t = blockIdx.x * blockDim.x + threadIdx.x;
  if (t >= E) return;
  int d = dst[t];
  float ex = expf(ebuf[t] - fdec(menc[d]));
  exbuf[t] = ex;
  atomicAdd(&denom[d], ex);
}

// one lane per (edge, feature); alpha optionally exported (layer-2 convs)
__global__ void k_edge_agg(const int* __restrict__ src, const int* __restrict__ dst,
                           const float* __restrict__ exbuf, const float* __restrict__ denom,
                           const float* __restrict__ hs, float* __restrict__ out,
                           float* __restrict__ alpha_out, int E) {
  long gid = (long)blockIdx.x * blockDim.x + threadIdx.x;
  int e = (int)(gid >> 6);
  if (e >= E) return;
  int f = (int)(gid & 63);
  int s = src[e], d = dst[e];
  float alpha = exbuf[e] / (denom[d] + 1e-16f);
  if (alpha_out && f == 0) alpha_out[e] = alpha;
  atomicAdd(&out[(size_t)d * HH + f], alpha * hs[(size_t)s * HH + f]);
}

// ================= host orchestration =================
extern "C" void kernel_launch(void* const* d_in, const int* in_sizes, int n_in,
                              void* d_out, int out_size, void* d_ws, size_t ws_size,
                              hipStream_t stream) {
  // inputs in setup_inputs() insertion order (nested dicts flattened in order)
  const float* x_user = (const float*)d_in[0];   // 200000 x 32
  const float* x_item = (const float*)d_in[1];   // 100000 x 32
  const int* eidx = (const int*)d_in[2];         // 2 x E
  const int* lidx = (const int*)d_in[3];         // 2 x L
  const int *src_u = eidx, *dst_i = eidx + EE;
  const int *lrow = lidx, *lcol = lidx + LL;
  // conv params: W_src, W_dst, att_src, att_dst, b
  const float *c1u_Ws = (const float*)d_in[4],  *c1u_Wd = (const float*)d_in[5],
              *c1u_as = (const float*)d_in[6],  *c1u_ad = (const float*)d_in[7],
              *c1u_b  = (const float*)d_in[8];
  const float *c1i_Ws = (const float*)d_in[9],  *c1i_Wd = (const float*)d_in[10],
              *c1i_as = (const float*)d_in[11], *c1i_ad = (const float*)d_in[12],
              *c1i_b  = (const float*)d_in[13];
  const float *l1u_W = (const float*)d_in[14], *l1u_b = (const float*)d_in[15];
  const float *l1i_W = (const float*)d_in[16], *l1i_b = (const float*)d_in[17];
  const float *c2u_Ws = (const float*)d_in[18], *c2u_Wd = (const float*)d_in[19],
              *c2u_as = (const float*)d_in[20], *c2u_ad = (const float*)d_in[21],
              *c2u_b  = (const float*)d_in[22];
  const float *c2i_Ws = (const float*)d_in[23], *c2i_Wd = (const float*)d_in[24],
              *c2i_as = (const float*)d_in[25], *c2i_ad = (const float*)d_in[26],
              *c2i_b  = (const float*)d_in[27];
  const float *d1_W = (const float*)d_in[28], *d1_b = (const float*)d_in[29];
  const float *d2_W = (const float*)d_in[30], *d2_b = (const float*)d_in[31];

  float* out_z   = (float*)d_out;        // L
  float* out_a1  = out_z + LL;           // alpha_u2i, E
  float* out_a2  = out_z + LL + EE;      // alpha_i2u, E

  // workspace carve (floats)
  float* ws = (float*)d_ws;
  size_t off = 0;
  auto alloc = [&](size_t n) { float* p = ws + off; off += n; return p; };
  float* hsU = alloc((size_t)NU * HH);  // T aliases this zone later
  float* hsI = alloc((size_t)NI * HH);
  float* zU  = alloc((size_t)NU * HH);
  float* zI  = alloc((size_t)NI * HH);
  float* hU  = alloc((size_t)NU * HH);  // h_user / z2_user
  float* hI  = alloc((size_t)NI * HH);  // h_item / z2_item
  float* ebuf  = alloc(EE);
  float* exbuf = alloc(EE);
  float* aS = alloc(NU);
  float* aD = alloc(NU);
  unsigned* menc = (unsigned*)alloc(NU);
  float* denom = alloc(NU);
  float* wcomb = alloc(128);
  float* T = hsU;  // L*64 = 32M floats exactly == hsU+hsI+zU zone (all dead at decoder)

  auto run_conv = [&](const float* Asrc, int Ns, const float* Adst, int Nd, int K,
                      const float* Wsrc, const float* Wdst,
                      const float* attS, const float* attD, const float* bconv,
                      const int* sidx, const int* didx,
                      float* hs, float* outbuf, float* alpha_out) {
    // hs = Asrc @ Wsrc
    k_gemm_wmma<<<Ns / 32, 256, 0, stream>>>(Asrc, Wsrc, nullptr, nullptr, hs, K, 0);
    // a_src = hs . att_src
    k_dot_rows<<<(Ns + 7) / 8, 256, 0, stream>>>(hs, attS, aS, Ns, HH, 0, nullptr);
    // wcomb = Wdst @ att_dst  (K-vector);  a_dst = Adst . wcomb
    k_dot_rows<<<(K + 7) / 8, 256, 0, stream>>>(Wdst, attD, wcomb, K, HH, 0, nullptr);
    k_dot_rows<<<(Nd + 7) / 8, 256, 0, stream>>>(Adst, wcomb, aD, Nd, K, 0, nullptr);
    // init segment state + bias-seeded accumulator
    k_fill_u32<<<(Nd + 255) / 256, 256, 0, stream>>>(menc, 0u, Nd);
    k_fill_u32<<<(Nd + 255) / 256, 256, 0, stream>>>((unsigned*)denom, 0u, Nd);
    k_init_bias<<<(Nd * HH + 255) / 256, 256, 0, stream>>>(outbuf, bconv, Nd * HH);
    // edge phase
    k_edge_e_max<<<(EE + 255) / 256, 256, 0, stream>>>(sidx, didx, aS, aD, ebuf, menc, EE);
    k_edge_exp_sum<<<(EE + 255) / 256, 256, 0, stream>>>(didx, ebuf, menc, exbuf, denom, EE);
    k_edge_agg<<<(EE * 64) / 256, 256, 0, stream>>>(sidx, didx, exbuf, denom, hs, outbuf,
                                                    alpha_out, EE);
  };

  // ---- layer 1 ----
  run_conv(x_user, NU, x_item, NI, DIN, c1u_Ws, c1u_Wd, c1u_as, c1u_ad, c1u_b,
           src_u, dst_i, hsU, hI, nullptr);                       // h_item
  run_conv(x_item, NI, x_user, NU, DIN, c1i_Ws, c1i_Wd, c1i_as, c1i_ad, c1i_b,
           dst_i, src_u, hsI, hU, nullptr);                       // h_user
  // z = relu(h + x@W + b)
  k_gemm_wmma<<<NU / 32, 256, 0, stream>>>(x_user, l1u_W, l1u_b, hU, zU, DIN, 1);
  k_gemm_wmma<<<NI / 32, 256, 0, stream>>>(x_item, l1i_W, l1i_b, hI, zI, DIN, 1);

  // ---- layer 2 (alphas exported) ----
  run_conv(zU, NU, zI, NI, HH, c2u_Ws, c2u_Wd, c2u_as, c2u_ad, c2u_b,
           src_u, dst_i, hsU, hI, out_a1);                        // z2_item
  run_conv(zI, NI, zU, NU, HH, c2i_Ws, c2i_Wd, c2i_as, c2i_ad, c2i_b,
           dst_i, src_u, hsI, hU, out_a2);                        // z2_user

  // ---- decoder ----
  k_dec_gemm_wmma<<<LL / 32, 256, 0, stream>>>(hU, hI, lrow, lcol, d1_W, d1_b, T);
  k_dot_rows<<<(LL + 7) / 8, 256, 0, stream>>>(T, d2_W, out_z, LL, HH, 1, d2_b);
}